// BOWmodel_26310969655524
// MI455X (gfx1250) — compile-verified
//
#include <hip/hip_runtime.h>

typedef __attribute__((ext_vector_type(2))) float v2f;
typedef __attribute__((ext_vector_type(4))) float v4f;
typedef __attribute__((ext_vector_type(8))) float v8f;

#define BSZ   1024
#define LMAX  200
#define HDIM  512
#define NOUT  5
#define BN_EPS 1e-5f

// ---------------------------------------------------------------------------
// Kernel 1: ragged embedding-bag mean. One block per example, 128 threads,
// each thread owns 4 consecutive floats of the 512-wide row (16B coalesced
// loads, 2KB per gathered row across the block). Tokens staged in LDS.
// ---------------------------------------------------------------------------
__global__ __launch_bounds__(128) void embed_pool_kernel(
    const int* __restrict__ tokens, const int* __restrict__ lengths,
    const float* __restrict__ emb, float* __restrict__ pooled) {
  __shared__ int s_tok[LMAX];
  const int b = blockIdx.x;
  const int t = threadIdx.x;
  const int len = lengths[b];
  for (int i = t; i < len; i += 128) s_tok[i] = tokens[b * LMAX + i];
  __syncthreads();

  v4f acc = {0.f, 0.f, 0.f, 0.f};
  const int col = t * 4;
  int l = 0;
  // unroll-by-2 so two independent 16B gathers are in flight per iter
  for (; l + 1 < len; l += 2) {
    v4f v0 = *(const v4f*)(emb + (size_t)s_tok[l]     * HDIM + col);
    v4f v1 = *(const v4f*)(emb + (size_t)s_tok[l + 1] * HDIM + col);
    acc += v0;
    acc += v1;
  }
  if (l < len) acc += *(const v4f*)(emb + (size_t)s_tok[l] * HDIM + col);

  acc *= (1.0f / (float)len);
  *(v4f*)(pooled + (size_t)b * HDIM + col) = acc;
}

// ---------------------------------------------------------------------------
// Kernel 2: H = pooled @ W1 + b1 via native fp32 WMMA (V_WMMA_F32_16X16X4_F32).
// One wave per 16x16 output tile; 2048 tiles -> 256 blocks x 8 waves.
// Bias is folded into the accumulator init (all 8 C VGPRs of a lane share the
// same output column).
// ---------------------------------------------------------------------------
__global__ __launch_bounds__(256) void gemm1_wmma_kernel(
    const float* __restrict__ A,      // [1024,512] pooled
    const float* __restrict__ W1,     // [512,512] row-major (k-major rows)
    const float* __restrict__ b1,     // [512]
    float* __restrict__ Hout) {       // [1024,512]
  const int lane = threadIdx.x & 31;
  const int wave = (blockIdx.x * 256 + threadIdx.x) >> 5;   // 0..2047
  const int tilesN = HDIM / 16;                              // 32
  const int tm = wave / tilesN;                              // 0..63
  const int tn = wave % tilesN;                              // 0..31
  const int m0 = tm * 16, n0 = tn * 16;
  const int half = lane >> 4;   // K split {0,1} vs {2,3}
  const int l15  = lane & 15;
  const int col  = n0 + l15;

  const float bias = b1[col];
  v8f c = {bias, bias, bias, bias, bias, bias, bias, bias};

  // A: lane row = m0+l15, packed K pair at k + 2*half  -> one b64 load
  const float* aptr = A + (size_t)(m0 + l15) * HDIM + 2 * half;
  // B: b.x = W1[k+2*half][col], b.y = W1[k+2*half+1][col]
  const float* bptr = W1 + (size_t)(2 * half) * HDIM + col;

#pragma unroll 4
  for (int k = 0; k < HDIM; k += 4) {
    v2f a = *(const v2f*)(aptr + k);
    v2f b;
    b.x = bptr[(size_t)k * HDIM];
    b.y = bptr[(size_t)(k + 1) * HDIM];
    // 8 args: (neg_a, A, neg_b, B, c_mod, C, reuse_a, reuse_b)
    c = __builtin_amdgcn_wmma_f32_16x16x4_f32(
        false, a, false, b, (short)0, c, false, false);
  }

  const int row = m0 + 8 * half;   // C: VGPR r -> row m0+r / m0+8+r
#pragma unroll
  for (int r = 0; r < 8; ++r)
    Hout[(size_t)(row + r) * HDIM + col] = c[r];
}

// ---------------------------------------------------------------------------
// Kernel 3: batch-norm statistics over the batch axis. 8 blocks x 256 threads;
// a block owns 64 columns; threads form a 4(row)x64(col) grid so global loads
// are coalesced. Biased variance, rsig = rsqrt(var+eps).
// ---------------------------------------------------------------------------
__global__ __launch_bounds__(256) void bn_stats_kernel(
    const float* __restrict__ Hm, float* __restrict__ mu,
    float* __restrict__ rsig) {
  const int c0 = blockIdx.x * 64;
  const int tx = threadIdx.x & 63;   // column within block
  const int ty = threadIdx.x >> 6;   // 0..3
  float s = 0.f, s2 = 0.f;
  for (int r = ty; r < BSZ; r += 4) {
    float v = Hm[(size_t)r * HDIM + c0 + tx];
    s += v;
    s2 = fmaf(v, v, s2);
  }
  __shared__ float shs[4][64];
  __shared__ float shq[4][64];
  shs[ty][tx] = s;
  shq[ty][tx] = s2;
  __syncthreads();
  if (ty == 0) {
    float S  = shs[0][tx] + shs[1][tx] + shs[2][tx] + shs[3][tx];
    float S2 = shq[0][tx] + shq[1][tx] + shq[2][tx] + shq[3][tx];
    float m   = S * (1.0f / BSZ);
    float var = S2 * (1.0f / BSZ) - m * m;
    mu[c0 + tx]   = m;
    rsig[c0 + tx] = rsqrtf(var + BN_EPS);
  }
}

// ---------------------------------------------------------------------------
// Kernel 4: normalize + ReLU + (512 -> 5) projection + b2, fused.
// One block (256 threads) per batch row; LDS tree reduction of 5 partials.
// ---------------------------------------------------------------------------
__global__ __launch_bounds__(256) void bn_relu_fc2_kernel(
    const float* __restrict__ Hm, const float* __restrict__ mu,
    const float* __restrict__ rsig, const float* __restrict__ gamma,
    const float* __restrict__ beta, const float* __restrict__ W2,
    const float* __restrict__ b2, float* __restrict__ out) {
  const int b = blockIdx.x;
  const int t = threadIdx.x;
  float acc[NOUT] = {0.f, 0.f, 0.f, 0.f, 0.f};
  for (int h = t; h < HDIM; h += 256) {
    float v = (Hm[(size_t)b * HDIM + h] - mu[h]) * rsig[h] * gamma[h] + beta[h];
    v = fmaxf(v, 0.0f);
#pragma unroll
    for (int j = 0; j < NOUT; ++j)
      acc[j] = fmaf(v, W2[h * NOUT + j], acc[j]);
  }
  __shared__ float sh[NOUT][256];
#pragma unroll
  for (int j = 0; j < NOUT; ++j) sh[j][t] = acc[j];
  __syncthreads();
  for (int s = 128; s > 0; s >>= 1) {
    if (t < s) {
#pragma unroll
      for (int j = 0; j < NOUT; ++j) sh[j][t] += sh[j][t + s];
    }
    __syncthreads();
  }
  if (t < NOUT) out[(size_t)b * NOUT + t] = sh[t][0] + b2[t];
}

// ---------------------------------------------------------------------------
// Launch. Inputs (setup_inputs order): tokens, lengths, emb, W1, b1, gamma,
// beta, W2, b2. Output: float[1024,5]. Workspace layout:
//   [0, 2MB)      pooled  float[1024*512]
//   [2MB, 4MB)    h       float[1024*512]
//   [4MB, ...)    mu[512], rsig[512]
// ---------------------------------------------------------------------------
extern "C" void kernel_launch(void* const* d_in, const int* in_sizes, int n_in,
                              void* d_out, int out_size, void* d_ws, size_t ws_size,
                              hipStream_t stream) {
  const int*   tokens  = (const int*)d_in[0];
  const int*   lengths = (const int*)d_in[1];
  const float* emb     = (const float*)d_in[2];
  const float* W1      = (const float*)d_in[3];
  const float* b1      = (const float*)d_in[4];
  const float* gamma   = (const float*)d_in[5];
  const float* beta    = (const float*)d_in[6];
  const float* W2      = (const float*)d_in[7];
  const float* b2      = (const float*)d_in[8];
  float* out = (float*)d_out;

  char* ws = (char*)d_ws;
  float* pooled = (float*)(ws);
  float* hbuf   = (float*)(ws + (size_t)BSZ * HDIM * sizeof(float));
  float* mu     = (float*)(ws + 2 * (size_t)BSZ * HDIM * sizeof(float));
  float* rsig   = mu + HDIM;

  embed_pool_kernel<<<BSZ, 128, 0, stream>>>(tokens, lengths, emb, pooled);
  // 2048 tiles (64 M-tiles x 32 N-tiles), 8 waves/block -> 256 blocks
  gemm1_wmma_kernel<<<256, 256, 0, stream>>>(pooled, W1, b1, hbuf);
  bn_stats_kernel<<<HDIM / 64, 256, 0, stream>>>(hbuf, mu, rsig);
  bn_relu_fc2_kernel<<<BSZ, 256, 0, stream>>>(hbuf, mu, rsig, gamma, beta, W2,
                                              b2, out);
}